// MHCLayerAITERFused_20555713478908
// MI455X (gfx1250) — compile-verified
//
#include <hip/hip_runtime.h>
#include <hip/hip_bf16.h>
#include <math.h>

// ---------------------------------------------------------------------------
// MHC fused layer for MI455X (gfx1250, wave32).
//
// out[b,n,c] = sum_m M[n,m]*x[b,m,c] + 2*sigmoid(H_post[n]) * bf16rt( (x_agg/rms)*w[c] )
//   x_agg[b,c] = sum_n sigmoid(H_pre[n]) * x[b,n,c]
//   rms        = sqrt(mean_c(x_agg^2) + 1e-6)
//   M          = sinkhorn(exp(H_res), 3 iters)
//
// Memory-bound (~1.07 GB HBM traffic -> ~46 us floor at 23.3 TB/s): single
// fused pass. x row staged in LDS via global_load_async_to_lds_b128; the 4x4
// stream mix runs on the matrix pipe via V_WMMA_F32_16X16X4_F32 with the
// RMSNorm branch injected through the WMMA C accumulator.
// ---------------------------------------------------------------------------

typedef float v2f __attribute__((ext_vector_type(2)));
typedef float v4f __attribute__((ext_vector_type(4)));
typedef float v8f __attribute__((ext_vector_type(8)));

static constexpr int kB = 8192;
static constexpr int kN = 4;
static constexpr int kC = 4096;
static constexpr float kEps = 1e-6f;

// LDS layout (floats). x rows padded by 16 words so the phase-2 transposed
// gather (lanes 0-15 -> rows 0/1, lanes 16-31 -> rows 2/3) is bank-conflict free.
static constexpr int XS_STRIDE = kC + 16;                   // 4112
static constexpr int Y_OFF     = kN * XS_STRIDE;            // 16448: rounded y_norm row
static constexpr int TILE_OFF  = Y_OFF + kC;                // 20544
static constexpr int TILE_ROW  = 18;                        // 16 cols + pad (8B-aligned b64 reads)
static constexpr int TILE_WAVE = kN * TILE_ROW;             // 72 floats per buffer
static constexpr int RED_OFF   = TILE_OFF + 8 * 2 * TILE_WAVE; // ping-pong: 21696
static constexpr int LDS_FLOATS = RED_OFF + 12;             // ~85 KB total

// ---------------------------------------------------------------------------
// Prologue: tiny 4x4 Sinkhorn + activations -> d_ws[0..23]
//   [0..15]  M (row-major)   [16..19] sigmoid(H_pre)   [20..23] 2*sigmoid(H_post)
// ---------------------------------------------------------------------------
__global__ void mhc_precompute_kernel(const float* __restrict__ Hpre,
                                      const float* __restrict__ Hpost,
                                      const float* __restrict__ Hres,
                                      float* __restrict__ pc) {
    if (threadIdx.x != 0 || blockIdx.x != 0) return;
    float P[4][4];
    for (int i = 0; i < 4; ++i)
        for (int j = 0; j < 4; ++j) P[i][j] = expf(Hres[i * 4 + j]);
    for (int it = 0; it < 3; ++it) {
        for (int i = 0; i < 4; ++i) {
            float s = P[i][0] + P[i][1] + P[i][2] + P[i][3] + kEps;
            for (int j = 0; j < 4; ++j) P[i][j] /= s;
        }
        for (int j = 0; j < 4; ++j) {
            float s = P[0][j] + P[1][j] + P[2][j] + P[3][j] + kEps;
            for (int i = 0; i < 4; ++i) P[i][j] /= s;
        }
    }
    for (int i = 0; i < 4; ++i)
        for (int j = 0; j < 4; ++j) pc[i * 4 + j] = P[i][j];
    for (int n = 0; n < 4; ++n) pc[16 + n] = 1.0f / (1.0f + expf(-Hpre[n]));
    for (int n = 0; n < 4; ++n) pc[20 + n] = 2.0f / (1.0f + expf(-Hpost[n]));
}

__device__ __forceinline__ float bf16_round(float v) {
    return __bfloat162float(__float2bfloat16(v));
}

// ---------------------------------------------------------------------------
// Main fused kernel: one block (256 threads = 8 waves) per batch row b.
// ---------------------------------------------------------------------------
__global__ __launch_bounds__(256)
void mhc_fused_kernel(const float* __restrict__ x,
                      const float* __restrict__ rw,   // rmsnorm weight [4096]
                      const float* __restrict__ pc,   // precomputed 24 floats
                      float* __restrict__ out) {
    extern __shared__ float smem[];
    float* xs   = smem;                 // [4][XS_STRIDE] padded x row
    float* ylds = smem + Y_OFF;         // [4096] bf16-rounded y_norm row
    float* red  = smem + RED_OFF;       // reduction scratch

    const int t    = threadIdx.x;
    const int lane = t & 31;
    const int wid  = t >> 5;
    const int b    = blockIdx.x;

    const float* xb = x + (size_t)b * (kN * kC);
    float*       ob = out + (size_t)b * (kN * kC);

    // ---- Phase 0: async-copy the 64 KB x row into LDS (16 x b128 per thread)
    const unsigned smem_base = (unsigned)(uintptr_t)smem;  // low 32 bits = LDS offset
#pragma unroll
    for (int n = 0; n < kN; ++n) {
#pragma unroll
        for (int k = 0; k < 4; ++k) {
            const float* gp = xb + n * kC + k * 1024 + t * 4;
            unsigned la = smem_base + (unsigned)((n * XS_STRIDE + k * 1024 + t * 4) * 4);
            asm volatile("global_load_async_to_lds_b128 %0, %1, off"
                         :: "v"(la), "v"(gp) : "memory");
        }
    }
    // rmsnorm weight: 4 coalesced b128 global loads, issued now so the L2
    // latency hides behind the async wait + phase 1.
    v4f wreg[4];
#pragma unroll
    for (int k = 0; k < 4; ++k) wreg[k] = *(const v4f*)&rw[t * 4 + k * 1024];

    asm volatile("s_wait_asynccnt 0" ::: "memory");
    __syncthreads();   // other waves' async LDS writes

    // ---- Phase 1: x_agg kept in registers + sum-of-squares reduction ----
    const float sp0 = pc[16], sp1 = pc[17], sp2 = pc[18], sp3 = pc[19];
    v4f areg[4];
    float sumsq = 0.0f;
#pragma unroll
    for (int k = 0; k < 4; ++k) {
        const int c = t * 4 + k * 1024;
        v4f x0 = *(const v4f*)&xs[0 * XS_STRIDE + c];
        v4f x1 = *(const v4f*)&xs[1 * XS_STRIDE + c];
        v4f x2 = *(const v4f*)&xs[2 * XS_STRIDE + c];
        v4f x3 = *(const v4f*)&xs[3 * XS_STRIDE + c];
        v4f a  = sp0 * x0 + sp1 * x1 + sp2 * x2 + sp3 * x3;
        areg[k] = a;
        sumsq += a.x * a.x + a.y * a.y + a.z * a.z + a.w * a.w;
    }
#pragma unroll
    for (int off = 16; off > 0; off >>= 1) sumsq += __shfl_xor(sumsq, off, 32);
    if (lane == 0) red[wid] = sumsq;
    __syncthreads();
    if (t == 0) {
        float s = 0.0f;
        for (int i = 0; i < 8; ++i) s += red[i];
        red[8] = 1.0f / sqrtf(s * (1.0f / (float)kC) + kEps);
    }
    __syncthreads();
    const float invr = red[8];

    // ---- Phase 1.5: rounded y_norm row into LDS (no agg re-read, w already
    // in registers; one coalesced v4 LDS store per chunk) ----
#pragma unroll
    for (int k = 0; k < 4; ++k) {
        const int c = t * 4 + k * 1024;
        v4f y4;
        y4.x = bf16_round(areg[k].x * invr * wreg[k].x);
        y4.y = bf16_round(areg[k].y * invr * wreg[k].y);
        y4.z = bf16_round(areg[k].z * invr * wreg[k].z);
        y4.w = bf16_round(areg[k].w * invr * wreg[k].w);
        *(v4f*)&ylds[c] = y4;
    }
    __syncthreads();

    // ---- Phase 2: WMMA mix + norm inject + coalesced store ----
    // A (16x4 f32, loop-invariant): lanes 0-15 rows, VGPR0={K0|K2}, VGPR1={K1|K3}.
    const bool hi = (lane >= 16);
    const int  j  = lane & 15;
    v2f A; A.x = 0.0f; A.y = 0.0f;
    if (j < 4) {                       // only rows 0-3 of A are real (rest zero)
        const int kb = hi ? 2 : 0;
        A.x = pc[j * 4 + kb];
        A.y = pc[j * 4 + kb + 1];
    }
    const float hp0 = pc[20], hp1 = pc[21], hp2 = pc[22], hp3 = pc[23];
    const int m0 = hi ? 2 : 0;         // B (4x16): VGPR0={K0|K2}, VGPR1={K1|K3}
    const int m1 = hi ? 3 : 1;
    float* tile0 = smem + TILE_OFF + wid * (2 * TILE_WAVE);
    const int on = lane >> 3;          // restage read: row n
    const int oc = (lane & 7) * 2;     // restage read: 2 cols per lane

    for (int tt = 0; tt < 32; ++tt) {
        const int c0 = wid * 512 + tt * 16;
        float* tile = tile0 + (tt & 1) * TILE_WAVE;   // ping-pong, no cross-iter alias

        // B tile: x[m, c0+j] gathered from padded LDS rows (conflict-free).
        v2f Bv;
        Bv.x = xs[m0 * XS_STRIDE + c0 + j];
        Bv.y = xs[m1 * XS_STRIDE + c0 + j];

        // C accumulator: rows 0-3 = hp[n] * y_norm, rows 4-15 zero.
        float y = ylds[c0 + j];
        y = hi ? 0.0f : y;             // lanes 16-31 hold C rows 8-15 -> zero
        v8f Cm = {hp0 * y, hp1 * y, hp2 * y, hp3 * y, 0.0f, 0.0f, 0.0f, 0.0f};

        // D = M * x_tile + C   (rows 0-3 are out[b, n, c0..c0+15])
        v8f D = __builtin_amdgcn_wmma_f32_16x16x4_f32(
            false, A, false, Bv, (short)0, Cm, false, false);

        // Restage rows 0-3 (lanes 0-15) through LDS, then full-wave b64 store.
        if (!hi) {
            tile[0 * TILE_ROW + j] = D[0];
            tile[1 * TILE_ROW + j] = D[1];
            tile[2 * TILE_ROW + j] = D[2];
            tile[3 * TILE_ROW + j] = D[3];
        }
        __builtin_amdgcn_wave_barrier();   // DS pipe is in-order per wave
        float o0 = tile[on * TILE_ROW + oc];
        float o1 = tile[on * TILE_ROW + oc + 1];
        float2 o2; o2.x = o0; o2.y = o1;
        *reinterpret_cast<float2*>(ob + on * kC + c0 + oc) = o2;
    }
}

// ---------------------------------------------------------------------------
extern "C" void kernel_launch(void* const* d_in, const int* in_sizes, int n_in,
                              void* d_out, int out_size, void* d_ws, size_t ws_size,
                              hipStream_t stream) {
    (void)in_sizes; (void)n_in; (void)out_size; (void)ws_size;
    const float* x     = (const float*)d_in[0];
    const float* rw    = (const float*)d_in[1];
    const float* hpre  = (const float*)d_in[2];
    const float* hpost = (const float*)d_in[3];
    const float* hres  = (const float*)d_in[4];
    float* out = (float*)d_out;
    float* pc  = (float*)d_ws;   // 24 floats of precomputed constants

    mhc_precompute_kernel<<<1, 32, 0, stream>>>(hpre, hpost, hres, pc);
    mhc_fused_kernel<<<kB, 256, LDS_FLOATS * sizeof(float), stream>>>(x, rw, pc, out);
}